// QuantumLayer_8950711845852
// MI455X (gfx1250) — compile-verified
//
#include <hip/hip_runtime.h>
#include <math.h>

typedef __attribute__((ext_vector_type(2))) float v2f;
typedef __attribute__((ext_vector_type(8))) float v8f;

#define NQ   12
#define NL   6
#define DIM  4096      // 2^12 amplitudes
#define TPB  256       // 8 waves of 32

// State index for group g (4 qubits as M-axis h in [0,16), remaining 8 qubits r in [0,256))
// qubit 0 is the most-significant bit of the 12-bit index (reference reshape order).
__device__ __forceinline__ int stateIdx(int g, int h, int r) {
    if (g == 0) return h * 256 + r;                              // qubits 0-3
    if (g == 1) return (r >> 4) * 256 + h * 16 + (r & 15);       // qubits 4-7
    return r * 16 + h;                                           // qubits 8-11
}

__global__ __launch_bounds__(TPB)
void qsim_kernel(const float* __restrict__ x,
                 const float* __restrict__ params,
                 float* __restrict__ out)
{
    __shared__ float sRe[DIM];
    __shared__ float sIm[DIM];
    __shared__ float Gr [256];   // Re(G)  row-major [h][k]
    __shared__ float Gi [256];   // Im(G)
    __shared__ float GiN[256];   // -Im(G) (f32 WMMA has no A/B negate)
    __shared__ float uRe[NQ][4], uIm[NQ][4];   // fused 2x2 per qubit
    __shared__ float cw[NQ], sw[NQ];           // cos/sin(x/2)
    __shared__ float zAcc[NQ];

    const int t    = threadIdx.x;
    const int b    = blockIdx.x;
    const int lane = t & 31;
    const int wave = t >> 5;

    // ---------------- initial product state  RY(x_q)|0> ----------------
    if (t < NQ) {
        float a = 0.5f * x[b * NQ + t];
        float s, c;
        sincosf(a, &s, &c);
        cw[t] = c; sw[t] = s;
    }
    __syncthreads();
    {
        // j = t*16 + v : bits 11..4 of j come from t (qubits 0..7), bits 3..0 from v (qubits 8..11)
        float pref = 1.0f;
        #pragma unroll
        for (int q = 0; q < 8; ++q) {
            int bit = (t >> (7 - q)) & 1;
            pref *= bit ? sw[q] : cw[q];
        }
        #pragma unroll
        for (int v = 0; v < 16; ++v) {
            float p = pref;
            #pragma unroll
            for (int q = 8; q < 12; ++q) {
                int bit = (v >> (11 - q)) & 1;
                p *= bit ? sw[q] : cw[q];
            }
            sRe[t * 16 + v] = p;
            sIm[t * 16 + v] = 0.0f;
        }
    }

    // ---------------- 6 HEA layers ----------------
    for (int l = 0; l < NL; ++l) {
        __syncthreads();
        // fused U_q = RZ(g) * RY(b) * RX(a), one qubit per thread
        if (t < NQ) {
            const float* pp = params + l * 3 * NQ + 3 * t;
            float ha = 0.5f * pp[0], hb = 0.5f * pp[1], hg = 0.5f * pp[2];
            float sa, ca, sb, cb, sg, cg;
            sincosf(ha, &sa, &ca);
            sincosf(hb, &sb, &cb);
            sincosf(hg, &sg, &cg);
            // M = RY * RX
            float m00r =  cb * ca, m00i =  sb * sa;
            float m01r = -sb * ca, m01i = -cb * sa;
            float m10r =  sb * ca, m10i = -cb * sa;
            float m11r =  cb * ca, m11i = -sb * sa;
            // row 0 *= e^{-i hg} = (cg - i sg) ;  row 1 *= e^{+i hg}
            uRe[t][0] = cg * m00r + sg * m00i;  uIm[t][0] = cg * m00i - sg * m00r;
            uRe[t][1] = cg * m01r + sg * m01i;  uIm[t][1] = cg * m01i - sg * m01r;
            uRe[t][2] = cg * m10r - sg * m10i;  uIm[t][2] = cg * m10i + sg * m10r;
            uRe[t][3] = cg * m11r - sg * m11i;  uIm[t][3] = cg * m11i + sg * m11r;
        }
        __syncthreads();

        #pragma unroll
        for (int g = 0; g < 3; ++g) {
            // ---- build G = U⊗U⊗U⊗U (one entry per thread) ----
            {
                int h = t >> 4, k = t & 15;
                float pr = 1.0f, pi = 0.0f;
                #pragma unroll
                for (int j = 0; j < 4; ++j) {
                    int hb2 = (h >> (3 - j)) & 1;
                    int kb2 = (k >> (3 - j)) & 1;
                    int e = hb2 * 2 + kb2;
                    float ur = uRe[g * 4 + j][e], ui = uIm[g * 4 + j][e];
                    float nr = pr * ur - pi * ui;
                    float ni = pr * ui + pi * ur;
                    pr = nr; pi = ni;
                }
                Gr[t] = pr; Gi[t] = pi; GiN[t] = -pi;
            }
            __syncthreads();

            // ---- complex 16x16 @ 16x256 GEMM via V_WMMA_F32_16X16X4_F32 ----
            // wave handles N-tiles 2*wave and 2*wave+1 (column-disjoint -> no cross-wave hazard)
            const int row  = lane & 15;          // A: M index / B: N index / D: N index
            const int ksel = (lane >> 4) * 2;    // lanes 0-15 -> K,K+1 ; lanes 16-31 -> K+2,K+3
            #pragma unroll
            for (int tt = 0; tt < 2; ++tt) {
                const int rB = (wave * 2 + tt) * 16;
                v8f dr = {};
                v8f di = {};
                #pragma unroll
                for (int kc = 0; kc < 4; ++kc) {
                    const int kA = kc * 4 + ksel;
                    v2f ar, ai, an;
                    ar.x = Gr [row * 16 + kA];  ar.y = Gr [row * 16 + kA + 1];
                    ai.x = Gi [row * 16 + kA];  ai.y = Gi [row * 16 + kA + 1];
                    an.x = GiN[row * 16 + kA];  an.y = GiN[row * 16 + kA + 1];
                    const int i0 = stateIdx(g, kA,     rB + row);
                    const int i1 = stateIdx(g, kA + 1, rB + row);
                    v2f br, bi;
                    br.x = sRe[i0];  br.y = sRe[i1];
                    bi.x = sIm[i0];  bi.y = sIm[i1];
                    // Re(D) += Re(G)Re(S) - Im(G)Im(S) ; Im(D) += Re(G)Im(S) + Im(G)Re(S)
                    dr = __builtin_amdgcn_wmma_f32_16x16x4_f32(false, ar, false, br, (short)0, dr, false, false);
                    dr = __builtin_amdgcn_wmma_f32_16x16x4_f32(false, an, false, bi, (short)0, dr, false, false);
                    di = __builtin_amdgcn_wmma_f32_16x16x4_f32(false, ar, false, bi, (short)0, di, false, false);
                    di = __builtin_amdgcn_wmma_f32_16x16x4_f32(false, ai, false, br, (short)0, di, false, false);
                }
                // D layout: VGPR m holds M=m (lanes 0-15) / M=m+8 (lanes 16-31), N = lane&15
                const int hB = (lane >> 4) * 8;
                #pragma unroll
                for (int m = 0; m < 8; ++m) {
                    const int idx = stateIdx(g, hB + m, rB + row);
                    sRe[idx] = dr[m];
                    sIm[idx] = di[m];
                }
            }
            __syncthreads();
        }

        // ---- CNOT ladder q->q+1 for q=0..10  ==  single Gray-code gather ----
        float trg[16], tig[16];
        #pragma unroll
        for (int v = 0; v < 16; ++v) {
            int j = t * 16 + v;
            int src = j ^ (j >> 1);
            trg[v] = sRe[src];
            tig[v] = sIm[src];
        }
        __syncthreads();
        #pragma unroll
        for (int v = 0; v < 16; ++v) {
            int j = t * 16 + v;
            sRe[j] = trg[v];
            sIm[j] = tig[v];
        }
    }

    // ---------------- <Z_q> = sum_i |psi_i|^2 * (1 - 2*bit_q(i)) ----------------
    if (t < NQ) zAcc[t] = 0.0f;
    __syncthreads();

    float acc[NQ];
    #pragma unroll
    for (int q = 0; q < NQ; ++q) acc[q] = 0.0f;
    #pragma unroll
    for (int v = 0; v < 16; ++v) {
        int j = t * 16 + v;
        float p = sRe[j] * sRe[j] + sIm[j] * sIm[j];
        #pragma unroll
        for (int q = 0; q < NQ; ++q)
            acc[q] += ((j >> (11 - q)) & 1) ? -p : p;
    }
    #pragma unroll
    for (int q = 0; q < NQ; ++q) {
        float vq = acc[q];
        #pragma unroll
        for (int off = 16; off > 0; off >>= 1)
            vq += __shfl_xor(vq, off, 32);
        if (lane == 0) atomicAdd(&zAcc[q], vq);   // ds_add_f32
    }
    __syncthreads();
    if (t < NQ) out[b * NQ + t] = zAcc[t];
}

extern "C" void kernel_launch(void* const* d_in, const int* in_sizes, int n_in,
                              void* d_out, int out_size, void* d_ws, size_t ws_size,
                              hipStream_t stream) {
    const float* x      = (const float*)d_in[0];   // (512, 12) f32
    const float* params = (const float*)d_in[1];   // (6, 36)   f32
    float* out          = (float*)d_out;           // (512, 12) f32
    const int batch = in_sizes[0] / NQ;
    qsim_kernel<<<batch, TPB, 0, stream>>>(x, params, out);
}